// Attention1D_30262339567718
// MI455X (gfx1250) — compile-verified
//
#include <hip/hip_runtime.h>

// Problem constants (match reference)
#define D_IN 512
#define DK   64
#define DV   64
#define B_   8
#define S_   2048

typedef __attribute__((ext_vector_type(2))) float v2f;
typedef __attribute__((ext_vector_type(8))) float v8f;

// D = A(16x4) * B(4x16) + C, fp32 WMMA (CDNA5 V_WMMA_F32_16X16X4_F32)
static __device__ __forceinline__ v8f wmma_f32(v2f a, v2f b, v8f c) {
  return __builtin_amdgcn_wmma_f32_16x16x4_f32(
      /*neg_a=*/false, a, /*neg_b=*/false, b,
      /*c_mod=*/(short)0, c, /*reuse_a=*/false, /*reuse_b=*/false);
}

// ---------------------------------------------------------------------------
// Kernel 0: transpose the three 64x512 weights into [3][512][64] so the
// projection GEMM's B-operand fragments are coalesced 64B row segments.
// ---------------------------------------------------------------------------
__global__ void transpose_weights(const float* __restrict__ Wk,
                                  const float* __restrict__ Wq,
                                  const float* __restrict__ Wv,
                                  float* __restrict__ WT3) {
  int idx = blockIdx.x * 256 + threadIdx.x;
  if (idx >= 3 * D_IN * DK) return;
  int mat = idx >> 15;          // / (512*64)
  int rem = idx & 32767;
  int d = rem >> 6;             // 0..511
  int n = rem & 63;             // 0..63
  const float* W = (mat == 0) ? Wk : ((mat == 1) ? Wq : Wv);
  WT3[idx] = W[n * D_IN + d];
}

// ---------------------------------------------------------------------------
// Kernel 1: fused K/Q/V projection.  One wave handles 16 rows of one batch:
// C(16x64) x3 matrices = 12 accumulator tiles, K-loop over 512 in slices of 4.
// A-frag: lane m holds X[s0+m, k]; VGPR0={k0 | k0+2}, VGPR1={k0+1 | k0+3}.
// ---------------------------------------------------------------------------
__global__ __launch_bounds__(256) void qkv_kernel(
    const float* __restrict__ X, const float* __restrict__ WT3,
    const float* __restrict__ bk, const float* __restrict__ bq,
    const float* __restrict__ bv,
    float* __restrict__ Kb, float* __restrict__ Qb, float* __restrict__ Vb) {
  const int lane = threadIdx.x & 31;
  const int wave = threadIdx.x >> 5;
  const int wg   = blockIdx.x * 8 + wave;   // 0..1023 s-tiles
  const int b    = wg >> 7;                 // batch = tile / 128
  const int s0   = (wg & 127) << 4;         // 16-row tile within sequence
  const int m    = lane & 15;
  const int hi   = lane >> 4;               // 0: lanes 0-15, 1: lanes 16-31

  const float* xr = X + (size_t)(b * S_ + s0 + m) * D_IN + 2 * hi;

  v8f acc[3][4];
#pragma unroll
  for (int i = 0; i < 3; ++i)
#pragma unroll
    for (int j = 0; j < 4; ++j)
      acc[i][j] = (v8f){0.f, 0.f, 0.f, 0.f, 0.f, 0.f, 0.f, 0.f};

  for (int t = 0; t < D_IN / 4; ++t) {
    v2f a;                                  // A-frag of X tile, k-slice t
    a.x = xr[4 * t];                        // k = 4t + 2*hi
    a.y = xr[4 * t + 1];                    // k = 4t + 2*hi + 1
    const float* wbase = WT3 + ((size_t)(4 * t + 2 * hi)) * DK + m;
#pragma unroll
    for (int mtx = 0; mtx < 3; ++mtx) {
#pragma unroll
      for (int nt = 0; nt < 4; ++nt) {
        const float* wp = wbase + (size_t)mtx * D_IN * DK + nt * 16;
        v2f bf;                             // B-frag of W^T, coalesced rows
        bf.x = wp[0];                       // row 4t+2hi
        bf.y = wp[DK];                      // row 4t+2hi+1
        acc[mtx][nt] = wmma_f32(a, bf, acc[mtx][nt]);
      }
    }
  }

  const float* bias[3] = {bk, bq, bv};
  float* out[3] = {Kb, Qb, Vb};
#pragma unroll
  for (int mtx = 0; mtx < 3; ++mtx) {
#pragma unroll
    for (int nt = 0; nt < 4; ++nt) {
      float bs = bias[mtx][nt * 16 + m];    // bias depends only on column N
#pragma unroll
      for (int r = 0; r < 8; ++r) {
        // C layout: VGPR r -> row r (lanes 0-15) / row r+8 (lanes 16-31)
        out[mtx][(size_t)(b * S_ + s0 + r + 8 * hi) * DK + nt * 16 + m] =
            acc[mtx][nt][r] + bs;
      }
    }
  }
}

// ---------------------------------------------------------------------------
// Kernel 2: fused attention with softmax over the BATCH axis.
// Block = 8 waves; wave w owns batch b=w for one 16-row i-tile.
// Loop over j-tiles of 16: E-tile via 16 WMMAs, exchange the 8 batch values
// per (i,j) through LDS, exact softmax (no streaming stats needed since the
// full reduction axis lives in this block), then Y += A*V via 16 WMMAs.
// E (8x2048x2048, 128MB) is never materialized.
// ---------------------------------------------------------------------------
__global__ __launch_bounds__(256) void attn_kernel(
    const float* __restrict__ Kb, const float* __restrict__ Qb,
    const float* __restrict__ Vb, float* __restrict__ Y) {
  // [r*32+lane][b], stride 12 dwords: 48B rows keep float4 reads 16B-aligned
  __shared__ float lds_e[256 * 12];
  // per-wave 16x16 A tile, stride 17 to avoid 64-bank conflicts
  __shared__ float lds_a[8][16 * 17];

  const int lane = threadIdx.x & 31;
  const int b    = threadIdx.x >> 5;        // wave id == batch index
  const int m    = lane & 15;
  const int hi   = lane >> 4;
  const int i0   = blockIdx.x << 4;         // 16-row i-tile

  const float scale = 0.125f;               // 1/sqrt(64), folded into K frags

  // Preload & pre-scale K A-fragments for this (b, i-tile): 16 slices x v2f
  v2f ka[16];
  {
    const float* kr = Kb + (size_t)(b * S_ + i0 + m) * DK + 2 * hi;
#pragma unroll
    for (int t = 0; t < 16; ++t) {
      ka[t].x = kr[4 * t] * scale;
      ka[t].y = kr[4 * t + 1] * scale;
    }
  }

  v8f yacc[4];
#pragma unroll
  for (int nt = 0; nt < 4; ++nt)
    yacc[nt] = (v8f){0.f, 0.f, 0.f, 0.f, 0.f, 0.f, 0.f, 0.f};

  for (int j0 = 0; j0 < S_; j0 += 16) {
    // ---- E tile: K(16x64) * Q^T(64x16).  B-frag of Q^T gathers like an
    // A-frag of Q (lane m holds row j0+m), adjacent pairs -> b64 loads.
    v8f e = (v8f){0.f, 0.f, 0.f, 0.f, 0.f, 0.f, 0.f, 0.f};
    {
      const float* qr = Qb + (size_t)(b * S_ + j0 + m) * DK + 2 * hi;
#pragma unroll
      for (int t = 0; t < 16; ++t) {
        v2f qb;
        qb.x = qr[4 * t];
        qb.y = qr[4 * t + 1];
        e = wmma_f32(ka[t], qb, e);
      }
    }

    // ---- cross-wave (cross-batch) exchange of E values
#pragma unroll
    for (int r = 0; r < 8; ++r)
      lds_e[(r * 32 + lane) * 12 + b] = e[r];
    __syncthreads();

    // ---- exact softmax over the 8 batch values per (i,j)
    float aval[8];
#pragma unroll
    for (int r = 0; r < 8; ++r) {
      const float4* p = (const float4*)&lds_e[(r * 32 + lane) * 12];
      float4 u = p[0];
      float4 w = p[1];
      float mx = fmaxf(fmaxf(fmaxf(u.x, u.y), fmaxf(u.z, u.w)),
                       fmaxf(fmaxf(w.x, w.y), fmaxf(w.z, w.w)));
      float z = __expf(u.x - mx) + __expf(u.y - mx) + __expf(u.z - mx) +
                __expf(u.w - mx) + __expf(w.x - mx) + __expf(w.y - mx) +
                __expf(w.z - mx) + __expf(w.w - mx);
      aval[r] = __expf(e[r] - mx) / z;      // own value == slot b of the 8
    }

    // ---- re-layout A from C-layout to A-operand layout via per-wave LDS
#pragma unroll
    for (int r = 0; r < 8; ++r)
      lds_a[b][(r + 8 * hi) * 17 + m] = aval[r];
    __syncthreads();   // also protects lds_e for the next iteration

    v2f af[4];
#pragma unroll
    for (int t = 0; t < 4; ++t) {
      af[t].x = lds_a[b][m * 17 + 4 * t + 2 * hi];
      af[t].y = lds_a[b][m * 17 + 4 * t + 2 * hi + 1];
    }

    // ---- Y(16x64) += A(16x16) * V(16x64); V B-frags are coalesced 64B rows
#pragma unroll
    for (int t = 0; t < 4; ++t) {
      const float* vr = Vb + (size_t)(b * S_ + j0 + 4 * t + 2 * hi) * DV + m;
#pragma unroll
      for (int nt = 0; nt < 4; ++nt) {
        v2f vb;
        vb.x = vr[nt * 16];                 // row j0+4t+2hi
        vb.y = vr[DV + nt * 16];            // row j0+4t+2hi+1
        yacc[nt] = wmma_f32(af[t], vb, yacc[nt]);
      }
    }
  }

  // ---- write Y tile (C layout -> (B,S,64))
#pragma unroll
  for (int nt = 0; nt < 4; ++nt)
#pragma unroll
    for (int r = 0; r < 8; ++r)
      Y[(size_t)(b * S_ + i0 + r + 8 * hi) * DV + nt * 16 + m] =
          yacc[nt][r];
}

// ---------------------------------------------------------------------------
extern "C" void kernel_launch(void* const* d_in, const int* in_sizes, int n_in,
                              void* d_out, int out_size, void* d_ws,
                              size_t ws_size, hipStream_t stream) {
  const float* X  = (const float*)d_in[0];
  const float* Wk = (const float*)d_in[1];
  const float* bk = (const float*)d_in[2];
  const float* Wq = (const float*)d_in[3];
  const float* bq = (const float*)d_in[4];
  const float* Wv = (const float*)d_in[5];
  const float* bv = (const float*)d_in[6];
  float* Y = (float*)d_out;

  // Workspace: K,Q,V (4MB each) + transposed weights (3*128KB) = ~12.4MB
  float* ws = (float*)d_ws;
  const size_t NKQV = (size_t)B_ * S_ * DK;   // 1,048,576 floats
  float* Kb  = ws;
  float* Qb  = ws + NKQV;
  float* Vb  = ws + 2 * NKQV;
  float* WT3 = ws + 3 * NKQV;                 // [3][512][64]

  transpose_weights<<<(3 * D_IN * DK + 255) / 256, 256, 0, stream>>>(
      Wk, Wq, Wv, WT3);
  qkv_kernel<<<(B_ * S_ / 16) / 8, 256, 0, stream>>>(
      X, WT3, bk, bq, bv, Kb, Qb, Vb);
  attn_kernel<<<S_ / 16, 256, 0, stream>>>(Kb, Qb, Vb, Y);
}